// RectangularMixedLoss_88115549044894
// MI455X (gfx1250) — compile-verified
//
#include <hip/hip_runtime.h>

// ---------------------------------------------------------------------------
// RectangularMixedLoss for MI455X (gfx1250, wave32).
// Memory-bound gather/scatter loss. Edge pass does both message directions per
// edge (halves edge_index/edge_attr traffic vs. the reference's concat) and
// scatters with global_atomic_add_f32 into an L2-resident (N,2) accumulator.
// Node pass fuses all three loss terms; wave reductions use
// V_WMMA_F32_16X16X4_F32 with an all-ones B matrix (layout-invariant), which
// sums p_m + p_{m+16} into every D column per the documented A / C-D layouts.
// ---------------------------------------------------------------------------

#define EPSF 1e-7f
#define N_ACC 6  // [0]=mse_num [1]=mse_den [2]=phys_sum [3]=pv_sum [4]=pv_cnt

typedef __attribute__((ext_vector_type(2))) float v2f;
typedef __attribute__((ext_vector_type(8))) float v8f;

// Full wave32 sum via one WMMA. Must be called with all 32 lanes active.
// A-layout (ISA 7.12.2, 32-bit 16x4): lane L<16 : VGPR0=A[L][0], VGPR1=A[L][1]
//                                     lane L>=16: VGPR0=A[L-16][2], VGPR1=A[L-16][3]
// We put each lane's partial in its VGPR0 slot, zero in VGPR1, B = ones(4x16)
// (all-ones is independent of B's lane layout). Then D[m][n] = x_m + x_{m+16}
// for every n; a lane's 8 D VGPRs are rows {m, m+1..} of its column, so the
// per-lane sum over d.s0..s7 gives sum_{m=0..7} for lanes<16 and sum_{m=8..15}
// for lanes>=16. xor-16 shuffle completes the total, broadcast to all lanes.
__device__ __forceinline__ float wave_reduce_sum(float x) {
  v2f a; a.x = x;    a.y = 0.0f;
  v2f b; b.x = 1.0f; b.y = 1.0f;
  v8f c = {};
  v8f d = __builtin_amdgcn_wmma_f32_16x16x4_f32(
      /*neg_a=*/false, a, /*neg_b=*/false, b,
      /*c_mod=*/(short)0, c, /*reuse_a=*/false, /*reuse_b=*/false);
  float s = d.s0 + d.s1 + d.s2 + d.s3 + d.s4 + d.s5 + d.s6 + d.s7;
  s += __shfl_xor(s, 16, 32);
  return s;
}

// ---------------------------------------------------------------------------
__global__ void zero_ws_kernel(unsigned int* __restrict__ ws, long long nwords) {
  long long i = (long long)blockIdx.x * blockDim.x + threadIdx.x;
  long long stride = (long long)gridDim.x * blockDim.x;
  for (; i < nwords; i += stride) ws[i] = 0u;
}

// ---------------------------------------------------------------------------
// One thread per edge; emit both directions (reference concatenates reversed
// edge list). Scatter with f32 atomics; calc_pq (8 MB) lives in L2.
__global__ void edge_flow_kernel(const float* __restrict__ pred,     // [N,2]
                                 const int*   __restrict__ eidx,     // [2,E]
                                 const float* __restrict__ eattr,    // [E,2]
                                 const float* __restrict__ emean,    // [1,2]
                                 const float* __restrict__ estd,     // [1,2]
                                 float* __restrict__ calc,           // [N,2] ws
                                 int E) {
  int e = blockIdx.x * blockDim.x + threadIdx.x;
  if (e >= E) return;

  float g = eattr[2 * e + 0] * (estd[0] + EPSF) + emean[0];
  float b = eattr[2 * e + 1] * (estd[1] + EPSF) + emean[1];

  int i = eidx[e];
  int j = eidx[E + e];

  float ei = pred[2 * i], fi = pred[2 * i + 1];
  float ej = pred[2 * j], fj = pred[2 * j + 1];

  float term1 = ei * ej + fi * fj;
  float term2 = fi * ej - ei * fj;   // direction i<-j; reversed is -term2
  float vsi   = ei * ei + fi * fi;
  float vsj   = ej * ej + fj * fj;

  // message targeting node i (x_i = node i, x_j = node j)
  float Pi =  g * (vsi - term1) - b * term2;
  float Qi = -b * (vsi - term1) - g * term2;
  // message targeting node j (roles swapped: term2 -> -term2, v_sq -> vsj)
  float Pj =  g * (vsj - term1) + b * term2;
  float Qj = -b * (vsj - term1) + g * term2;

  atomicAdd(&calc[2 * i + 0], Pi);
  atomicAdd(&calc[2 * i + 1], Qi);
  atomicAdd(&calc[2 * j + 0], Pj);
  atomicAdd(&calc[2 * j + 1], Qj);
}

// ---------------------------------------------------------------------------
// Fused node pass: masked MSE, physics residual, PV-bus voltage error.
// Fixed grid, grid-stride, so every thread reaches the WMMA reduction with
// EXEC all-ones. Lane 0 of each wave accumulates into f64 globals.
__global__ void node_loss_kernel(const float* __restrict__ pred,    // [N,2]
                                 const float* __restrict__ ty,      // [N,4]
                                 const float* __restrict__ mask,    // [N,4]
                                 const int*   __restrict__ bus,     // [N]
                                 const float* __restrict__ tvm,     // [N]
                                 const float* __restrict__ xymean,  // [1,4]
                                 const float* __restrict__ xystd,   // [1,4]
                                 const float* __restrict__ calc,    // [N,2] ws
                                 double* __restrict__ acc,          // [N_ACC] ws
                                 int N) {
  const float xs0 = xystd[0] + EPSF, xs1 = xystd[1] + EPSF;
  const float xs2 = xystd[2] + EPSF, xs3 = xystd[3] + EPSF;
  const float xm0 = xymean[0], xm1 = xymean[1];
  const float xm2 = xymean[2], xm3 = xymean[3];

  float mse_n = 0.0f, mse_d = 0.0f, phys = 0.0f, pv_s = 0.0f, pv_c = 0.0f;

  int stride = gridDim.x * blockDim.x;
  for (int v = blockIdx.x * blockDim.x + threadIdx.x; v < N; v += stride) {
    float p0 = pred[2 * v], p1 = pred[2 * v + 1];
    float t0 = ty[4 * v + 0], t1 = ty[4 * v + 1];
    float t2 = ty[4 * v + 2], t3 = ty[4 * v + 3];
    float m2 = mask[4 * v + 2], m3 = mask[4 * v + 3];

    // masked MSE on ef channels
    float d0 = p0 - t2, d1 = p1 - t3;
    mse_n += d0 * d0 * m2 + d1 * d1 * m3;
    mse_d += m2 + m3;

    // physics residual: (real_pq + calc_pq)^2
    float rp = t0 * xs0 + xm0;
    float rq = t1 * xs1 + xm1;
    float s0 = rp + calc[2 * v + 0];
    float s1 = rq + calc[2 * v + 1];
    phys += s0 * s0 + s1 * s1;

    // PV-bus voltage magnitude error
    float er = p0 * xs2 + xm2;
    float fr = p1 * xs3 + xm3;
    float vm2 = er * er + fr * fr;
    if (bus[v] == 1) {
      float t = tvm[v];
      pv_s += fabsf(vm2 - t * t);
      pv_c += 1.0f;
    }
  }

  // wave32 reductions via v_wmma_f32_16x16x4_f32 (EXEC all-ones here)
  mse_n = wave_reduce_sum(mse_n);
  mse_d = wave_reduce_sum(mse_d);
  phys  = wave_reduce_sum(phys);
  pv_s  = wave_reduce_sum(pv_s);
  pv_c  = wave_reduce_sum(pv_c);

  if ((threadIdx.x & 31) == 0) {
    atomicAdd(&acc[0], (double)mse_n);
    atomicAdd(&acc[1], (double)mse_d);
    atomicAdd(&acc[2], (double)phys);
    atomicAdd(&acc[3], (double)pv_s);
    atomicAdd(&acc[4], (double)pv_c);
  }
}

// ---------------------------------------------------------------------------
__global__ void finalize_kernel(const double* __restrict__ acc,
                                float* __restrict__ out, int N) {
  if (blockIdx.x == 0 && threadIdx.x == 0) {
    double mse  = acc[0] / (acc[1] + 1e-6);
    double phys = acc[2] / (2.0 * (double)N);
    double cnt  = acc[4];
    double pv   = (cnt > 0.0) ? (acc[3] / fmax(cnt, 1.0)) : 0.0;
    out[0] = (float)(0.8 * mse + 0.2 * phys + 0.1 * pv);
    out[1] = (float)mse;
    out[2] = (float)phys;
    out[3] = (float)pv;
  }
}

// ---------------------------------------------------------------------------
extern "C" void kernel_launch(void* const* d_in, const int* in_sizes, int n_in,
                              void* d_out, int out_size, void* d_ws, size_t ws_size,
                              hipStream_t stream) {
  const float* pred   = (const float*)d_in[0];   // (N,2)
  const float* ty     = (const float*)d_in[1];   // (N,4)
  const float* mask   = (const float*)d_in[2];   // (N,4)
  const int*   eidx   = (const int*)  d_in[3];   // (2,E)
  const float* eattr  = (const float*)d_in[4];   // (E,2)
  const int*   bus    = (const int*)  d_in[5];   // (N,)
  const float* tvm    = (const float*)d_in[6];   // (N,)
  const float* xymean = (const float*)d_in[7];   // (1,4)
  const float* xystd  = (const float*)d_in[8];   // (1,4)
  const float* emean  = (const float*)d_in[9];   // (1,2)
  const float* estd   = (const float*)d_in[10];  // (1,2)
  float* out = (float*)d_out;

  const int N = in_sizes[0] / 2;
  const int E = in_sizes[3] / 2;

  // workspace layout: [0, 2N) floats = calc_pq ; then N_ACC doubles (8B aligned
  // since 2N*4 bytes is a multiple of 8).
  float*  calc = (float*)d_ws;
  double* acc  = (double*)((char*)d_ws + (size_t)2 * N * sizeof(float));

  // 1) zero accumulators (calc_pq + scalar accumulators)
  long long nwords = (long long)2 * N + 2 * N_ACC;
  zero_ws_kernel<<<2048, 256, 0, stream>>>((unsigned int*)d_ws, nwords);

  // 2) edge scatter (both directions per edge)
  int eblocks = (E + 255) / 256;
  edge_flow_kernel<<<eblocks, 256, 0, stream>>>(pred, eidx, eattr, emean, estd,
                                                calc, E);

  // 3) fused node losses + WMMA wave reductions
  node_loss_kernel<<<2048, 256, 0, stream>>>(pred, ty, mask, bus, tvm,
                                             xymean, xystd, calc, acc, N);

  // 4) finalize scalars
  finalize_kernel<<<1, 32, 0, stream>>>(acc, out, N);
}